// Sampler_78726750536038
// MI455X (gfx1250) — compile-verified
//
#include <hip/hip_runtime.h>
#include <hip/hip_bf16.h>
#include <math.h>
#include <stdint.h>

// ---------------------------------------------------------------------------
// MI455X sampler kernel (gfx1250, wave32).
// One workgroup (512 threads = 16 wave32) per batch row.
// HBM-bound: ~67MB mandatory traffic -> ~2.9us floor at 23.3 TB/s.
// Logits (65.5MB) fit in the 192MB L2, so re-scan passes are L2-resident.
// Uses: LDS bitmask/hash/histogram/bitonic-sort, wave32 __shfl_xor reductions,
// global_prefetch_b8, and GLOBAL_LOAD_ASYNC_TO_LDS_B128 + s_wait_asynccnt
// for id staging (builtin type-checked and lowering confirmed in round 2).
// ---------------------------------------------------------------------------

#define NTHR   512
#define NWAVE  (NTHR / 32)
#define BMW    4096          // bitmask words: supports V <= 131072
#define HSZ    1024          // output-token count hash slots
#define CAP    1024          // candidate buffer capacity (need_n <= 256)
#define NB     1024          // histogram buckets over [-128, 128) step 0.25
#define HEMPTY 0xFFFFFFFFu
#define BIGNEG (-1.0e30f)

// GNU vector type matching the async-to-LDS builtin's parameter pointee
typedef int v4i_gnu __attribute__((vector_size(16)));

// Stage n ints (n multiple of 4 preferred) from global to LDS.
__device__ __forceinline__ void copy_ints_to_lds(unsigned* dst, const int* src,
                                                 int n, int tid) {
#if defined(__gfx1250__) && __has_builtin(__builtin_amdgcn_global_load_async_to_lds_b128)
  if ((((uintptr_t)src & 15u) == 0) && ((n & 3) == 0)) {
    for (int i = tid * 4; i < n; i += NTHR * 4) {
      __builtin_amdgcn_global_load_async_to_lds_b128(
          (v4i_gnu*)(src + i), (v4i_gnu*)(dst + i), 0, 0);
    }
    asm volatile("s_wait_asynccnt 0" ::: "memory");
    return;
  }
#endif
  for (int i = tid; i < n; i += NTHR) dst[i] = (unsigned)src[i];
}

__global__ __launch_bounds__(NTHR) void sampler_kernel(
    const float* __restrict__ logits, const int* __restrict__ prompt,
    const int* __restrict__ outtok, const int* __restrict__ stoptok,
    const int* __restrict__ mintok, const float* __restrict__ presp,
    const float* __restrict__ freqp, const float* __restrict__ repp,
    const float* __restrict__ temp, const int* __restrict__ topk,
    const float* __restrict__ topp, const float* __restrict__ noise,
    int* __restrict__ out_samp, float* __restrict__ out_lp,
    int* __restrict__ out_idx, int B, int V, int P, int O, int S, int K) {
  __shared__ unsigned bm[BMW];            // prompt|output token bitmask
  __shared__ unsigned hkey[HSZ];          // output-token hash keys
  __shared__ unsigned hcnt[HSZ];          // output-token hash counts
  __shared__ float    cval[CAP];          // candidate values
  __shared__ int      cidx[CAP];          // candidate vocab ids
  __shared__ int      hist[NB];           // top-tail histogram
  __shared__ int      seg[32];            // 32-bucket segment sums
  __shared__ __align__(16) unsigned stage[2048];
  __shared__ float red_m[NWAVE], red_s[NWAVE];
  __shared__ float red_bv[NWAVE];
  __shared__ int   red_bj[NWAVE];
  __shared__ int   sh_cnt;
  __shared__ float sh_T, sh_M, sh_Z;
  __shared__ int   sh_J;
  __shared__ float sh_x0, sh_logZk;
  __shared__ int   sh_fill[128];

  const int tid = threadIdx.x;
  const int r   = blockIdx.x;
  const float* lrow = logits + (size_t)r * (size_t)V;
  const float* nrow = noise  + (size_t)r * (size_t)V;

  // ----- per-row scalars -----
  const float rp     = repp[r];
  const float inv_rp = 1.0f / rp;
  const float fp     = freqp[r];
  const float pp     = presp[r];
  const float tr     = temp[r];
  const float teff   = (tr < 1e-5f) ? 1.0f : tr;
  const float invT   = 1.0f / teff;
  const bool  greedy = (tr < 1e-5f);
  const bool  need   = (mintok[r] > O);
  const float stop_pen = need ? BIGNEG : __builtin_inff();
  int stops[8];
  const int Sc = (S > 8) ? 8 : S;
  for (int s = 0; s < Sc; ++s) stops[s] = stoptok[(size_t)r * S + s];

  // ----- phase 0: init LDS -----
  for (int i = tid; i < BMW; i += NTHR) bm[i] = 0u;
  for (int i = tid; i < HSZ; i += NTHR) { hkey[i] = HEMPTY; hcnt[i] = 0u; }
  for (int i = tid; i < NB; i += NTHR) hist[i] = 0;
  if (tid == 0) sh_cnt = 0;
  __syncthreads();

  // ----- phase 1a: prompt token scatter (bit set) -----
  for (int base = 0; base < P; base += 2048) {
    const int n = (P - base) < 2048 ? (P - base) : 2048;
    copy_ints_to_lds(stage, prompt + (size_t)r * P + base, n, tid);
    __syncthreads();
    for (int i = tid; i < n; i += NTHR) {
      const unsigned v = stage[i];
      if ((v >> 5) < BMW) atomicOr(&bm[v >> 5], 1u << (v & 31u));
    }
    __syncthreads();
  }
  // ----- phase 1b: output token scatter (bit set + hash count) -----
  for (int base = 0; base < O; base += 2048) {
    const int n = (O - base) < 2048 ? (O - base) : 2048;
    copy_ints_to_lds(stage, outtok + (size_t)r * O + base, n, tid);
    __syncthreads();
    for (int i = tid; i < n; i += NTHR) {
      const unsigned v = stage[i];
      if ((v >> 5) < BMW) atomicOr(&bm[v >> 5], 1u << (v & 31u));
      unsigned h = (v * 2654435761u) & (HSZ - 1);
      for (;;) {
        const unsigned prev = atomicCAS(&hkey[h], HEMPTY, v);
        if (prev == HEMPTY || prev == v) { atomicAdd(&hcnt[h], 1u); break; }
        h = (h + 1) & (HSZ - 1);
      }
    }
    __syncthreads();
  }

  // fused penalty pipeline: stop-min -> repetition -> freq/presence -> 1/T
  auto applyx = [&](float x, int v) -> float {
#pragma unroll 4
    for (int s = 0; s < Sc; ++s)
      if (v == stops[s]) x = fminf(x, stop_pen);
    const unsigned vv = (unsigned)v;
    const unsigned w = bm[vv >> 5];
    if ((w >> (vv & 31u)) & 1u) {
      x = (x > 0.0f) ? x * inv_rp : x * rp;
      unsigned h = (vv * 2654435761u) & (HSZ - 1);
      unsigned c = 0;
      for (;;) {
        const unsigned k = hkey[h];
        if (k == vv) { c = hcnt[h]; break; }
        if (k == HEMPTY) break;
        h = (h + 1) & (HSZ - 1);
      }
      if (c) x -= fp * (float)c + pp;
    }
    return x * invT;
  };

  // ----- phase 2: streaming online softmax + top-tail histogram -----
  float m_loc = -__builtin_inff(), s_loc = 0.0f;
  const int V4 = ((V & 3) == 0) ? (V >> 2) : 0;
  const float4* lrow4 = (const float4*)lrow;
  for (int q = tid; q < V4; q += NTHR) {
    const float4 Lv = lrow4[q];
    const int qp = q + NTHR * 4;
    if (qp < V4) __builtin_prefetch(&lrow4[qp], 0, 1);  // global_prefetch_b8
    const float xs[4] = {Lv.x, Lv.y, Lv.z, Lv.w};
#pragma unroll
    for (int e = 0; e < 4; ++e) {
      const int v = (q << 2) + e;
      const float x = applyx(xs[e], v);
      if (x > m_loc) { s_loc = s_loc * __expf(m_loc - x) + 1.0f; m_loc = x; }
      else            s_loc += __expf(x - m_loc);
      if (x > m_loc - 64.0f) {  // only the tail near max can matter
        int b = (int)floorf((x + 128.0f) * 4.0f);
        b = b < 0 ? 0 : (b > NB - 1 ? NB - 1 : b);
        atomicAdd(&hist[b], 1);
      }
    }
  }
  for (int v = (V4 << 2) + tid; v < V; v += NTHR) {  // scalar tail
    const float x = applyx(lrow[v], v);
    if (x > m_loc) { s_loc = s_loc * __expf(m_loc - x) + 1.0f; m_loc = x; }
    else            s_loc += __expf(x - m_loc);
    if (x > m_loc - 64.0f) {
      int b = (int)floorf((x + 128.0f) * 4.0f);
      b = b < 0 ? 0 : (b > NB - 1 ? NB - 1 : b);
      atomicAdd(&hist[b], 1);
    }
  }
  // wave32 shuffle reduction of (m, s)
  for (int off = 16; off > 0; off >>= 1) {
    const float m2 = __shfl_xor(m_loc, off, 32);
    const float s2 = __shfl_xor(s_loc, off, 32);
    if (m2 > m_loc) { s_loc = s_loc * __expf(m_loc - m2) + s2; m_loc = m2; }
    else            { s_loc = s_loc + s2 * __expf(m2 - m_loc); }
  }
  if ((tid & 31) == 0) { red_m[tid >> 5] = m_loc; red_s[tid >> 5] = s_loc; }
  __syncthreads();

  // parallel 32-bucket segment sums for the histogram threshold search
  if (tid < 32) {
    int ssum = 0;
    const int base = tid * 32;
#pragma unroll
    for (int k = 0; k < 32; ++k) ssum += hist[base + k];
    seg[tid] = ssum;
  }
  __syncthreads();

  if (tid == 0) {
    float M = red_m[0], Zs = red_s[0];
    for (int w = 1; w < NWAVE; ++w) {
      const float m2 = red_m[w], s2 = red_s[w];
      if (m2 > M) { Zs = Zs * __expf(M - m2) + s2; M = m2; }
      else        { Zs += s2 * __expf(m2 - M); }
    }
    sh_M = M; sh_Z = Zs;
    // threshold: need >= top-64 exact (cap at 256 so cnt stays << CAP)
    const int tk = topk[r];
    long long keff = (tk < 1) ? (long long)V : (long long)tk;
    int need_n = (int)(keff > 256 ? 256 : keff);
    if (need_n < 64) need_n = 64;
    if (need_n > V)  need_n = V;
    // two-level search: segments first, then <=32 buckets inside one segment
    int bsel = -1;
    int acc = 0;
    for (int t = 31; t >= 0; --t) {
      if (acc + seg[t] >= need_n) {
        int cum = acc;
        const int base = t * 32;
        for (int k = 31; k >= 0; --k) {
          cum += hist[base + k];
          if (cum >= need_n) { bsel = base + k; break; }
        }
        break;
      }
      acc += seg[t];
    }
    float T;
    if (bsel > 0)       T = (float)bsel * 0.25f - 128.0f;
    else if (bsel == 0) T = -3.0e38f;
    else                T = M - 64.0f;  // histogram complete above M-64
    sh_T = T;
  }
  __syncthreads();

  // ----- phase 3: collect candidates >= T (L2-resident re-scan) -----
  const float T = sh_T;
  for (int q = tid; q < V4; q += NTHR) {
    const float4 Lv = lrow4[q];
    const float xs[4] = {Lv.x, Lv.y, Lv.z, Lv.w};
#pragma unroll
    for (int e = 0; e < 4; ++e) {
      const int v = (q << 2) + e;
      const float x = applyx(xs[e], v);
      if (x >= T) {
        const int p = atomicAdd(&sh_cnt, 1);
        if (p < CAP) { cval[p] = x; cidx[p] = v; }
      }
    }
  }
  for (int v = (V4 << 2) + tid; v < V; v += NTHR) {
    const float x = applyx(lrow[v], v);
    if (x >= T) {
      const int p = atomicAdd(&sh_cnt, 1);
      if (p < CAP) { cval[p] = x; cidx[p] = v; }
    }
  }
  __syncthreads();
  int cnt = sh_cnt; if (cnt > CAP) cnt = CAP;
  for (int i = cnt + tid; i < CAP; i += NTHR) {
    cval[i] = -__builtin_inff(); cidx[i] = 0x7FFFFFFF;
  }
  __syncthreads();

  // ----- phase 4: bitonic sort, descending value / ascending index ties
  //       (matches JAX stable argsort(-x)); CAP=1024 -> 55 stages -----
  for (int k = 2; k <= CAP; k <<= 1) {
    for (int j = k >> 1; j > 0; j >>= 1) {
      for (int i = tid; i < CAP; i += NTHR) {
        const int ixj = i ^ j;
        if (ixj > i) {
          const bool up = ((i & k) == 0);
          const float a = cval[i], b2 = cval[ixj];
          const int ia = cidx[i], ib = cidx[ixj];
          const bool good = (a > b2) || (a == b2 && ia < ib);
          if (up ? !good : good) {
            cval[i] = b2; cval[ixj] = a;
            cidx[i] = ib; cidx[ixj] = ia;
          }
        }
      }
      __syncthreads();
    }
  }

  // ----- phase 5: prefix length J (top-k AND top-p), log-softmax consts -----
  if (tid == 0) {
    const float M = sh_M, Z = sh_Z;
    const float tp = topp[r];
    const int tk = topk[r];
    const long long keff = (tk < 1) ? (long long)V : (long long)tk;
    int lim = cnt;
    if ((long long)lim > keff) lim = (int)keff;
    float cum = 0.0f;
    int J = 0;
    for (int j = 0; j < lim; ++j) {
      if (cum < tp) { cum += __expf(cval[j] - M) / Z; ++J; }
      else break;
    }
    if (J < 1 && lim > 0) J = 1;
    const float x0 = (J > 0) ? cval[0] : 0.0f;
    float Zk = 0.0f;
    for (int j = 0; j < J; ++j) Zk += __expf(cval[j] - x0);
    sh_J = J; sh_x0 = x0; sh_logZk = logf(Zk > 0.0f ? Zk : 1.0f);
    // filler indices for top-K slots beyond J: smallest ids not kept
    int nf = K - J;
    if (nf > 0) {
      if (nf > 128) nf = 128;
      int c = 0;
      for (int v = 0; c < nf && v < V; ++v) {
        bool kept = false;
        for (int j = 0; j < J; ++j)
          if (cidx[j] == v) { kept = true; break; }
        if (!kept) sh_fill[c++] = v;
      }
      for (; c < nf; ++c) sh_fill[c] = 0;
    }
  }
  __syncthreads();
  const int J = sh_J;

  // ----- phase 6: Gumbel sampling over kept set (gather <=1024 noise vals) ---
  float bestv = -__builtin_inff();
  int   bestj = 0x7FFFFFFF;
  for (int j = tid; j < J; j += NTHR) {
    const float u = nrow[cidx[j]];
    const float g = -logf(-logf(u));
    const float sc = cval[j] * invT + g;   // reference divides by T twice
    if (sc > bestv) { bestv = sc; bestj = j; }
  }
  for (int off = 16; off > 0; off >>= 1) {
    const float v2 = __shfl_xor(bestv, off, 32);
    const int   j2 = __shfl_xor(bestj, off, 32);
    if (v2 > bestv || (v2 == bestv && j2 < bestj)) { bestv = v2; bestj = j2; }
  }
  if ((tid & 31) == 0) { red_bv[tid >> 5] = bestv; red_bj[tid >> 5] = bestj; }
  __syncthreads();
  if (tid == 0) {
    float bv = red_bv[0]; int bj = red_bj[0];
    for (int w = 1; w < NWAVE; ++w) {
      if (red_bv[w] > bv || (red_bv[w] == bv && red_bj[w] < bj)) {
        bv = red_bv[w]; bj = red_bj[w];
      }
    }
    int samp = 0;
    if (J > 0) samp = (greedy || bj >= J) ? cidx[0] : cidx[bj];
    out_samp[r] = samp;
  }

  // ----- phase 7: top-K logprobs + indices -----
  const float x0 = sh_x0, lZk = sh_logZk;
  for (int i = tid; i < K; i += NTHR) {
    float lp; int ix;
    if (i < J) { lp = cval[i] - x0 - lZk; ix = cidx[i]; }
    else {
      lp = BIGNEG - x0 - lZk;
      const int f = (i - J) < 128 ? (i - J) : 127;
      ix = sh_fill[f];
    }
    out_lp[(size_t)r * K + i] = lp;
    out_idx[(size_t)r * K + i] = ix;
  }
}

extern "C" void kernel_launch(void* const* d_in, const int* in_sizes, int n_in,
                              void* d_out, int out_size, void* d_ws,
                              size_t ws_size, hipStream_t stream) {
  (void)n_in; (void)d_ws; (void)ws_size;
  const float* logits = (const float*)d_in[0];
  const int*   prompt = (const int*)d_in[1];
  const int*   outtok = (const int*)d_in[2];
  const int*   stoptk = (const int*)d_in[3];
  const int*   mintok = (const int*)d_in[4];
  const float* presp  = (const float*)d_in[5];
  const float* freqp  = (const float*)d_in[6];
  const float* repp   = (const float*)d_in[7];
  const float* tempr  = (const float*)d_in[8];
  const int*   topk   = (const int*)d_in[9];
  const float* topp   = (const float*)d_in[10];
  const float* noise  = (const float*)d_in[11];

  const int B = in_sizes[4];
  if (B <= 0) return;
  const int V = in_sizes[0] / B;
  const int P = in_sizes[1] / B;
  const int O = in_sizes[2] / B;
  const int S = in_sizes[3] / B;
  const int K = (out_size > B) ? (out_size - B) / (2 * B) : 0;

  float* outf = (float*)d_out;
  int*   out_samp = (int*)outf;                       // B ints (bit-stored)
  float* out_lp   = outf + B;                         // B*K floats
  int*   out_idx  = (int*)(outf + B + (size_t)B * K); // B*K ints (bit-stored)

  sampler_kernel<<<B, NTHR, 0, stream>>>(
      logits, prompt, outtok, stoptk, mintok, presp, freqp, repp, tempr, topk,
      topp, noise, out_samp, out_lp, out_idx, B, V, P, O, S, K);
}